// HashEmbedder_52647709114928
// MI455X (gfx1250) — compile-verified
//
#include <hip/hip_runtime.h>
#include <hip/hip_bf16.h>

// Instant-NGP style hash-grid embedder for MI455X (gfx1250, wave32).
//
// Key observations vs. the reference:
//  * The reference truncates the 32-wide feature vector to 16 columns,
//    so only levels 0..7 contribute to the output. We skip levels 8..15.
//  * The 8 live tables total 32 MB -> fully resident in the 192 MB L2.
//    Gathers are L2 hits; keep them as regular (RT temporal) loads.
//  * The 64 MB output is write-once: use non-temporal B128 stores
//    (th:TH_STORE_NT) so it does not evict the tables from L2.
//  * Per-level structure (vs. an all-levels-batched gather phase) keeps
//    register pressure low -> higher occupancy; measured instruction mix
//    (620 total / 376 valu) beats the batched variant (842 / 563).
//  * Hash uses PRIMES = {1, 2654435761, 805459861}; since P0 == 1 and
//    (b+1)*P == b*P + P (mod 2^32), the six axis hashes cost 2 muls + 3 adds.

typedef float v4f __attribute__((ext_vector_type(4)));

#define NLV_OUT 8                    // levels that survive the [:, :16] truncation
#define LOG2_T  19
#define TSIZE   (1u << LOG2_T)
#define TMASK   (TSIZE - 1u)
#define P1      2654435761u
#define P2      805459861u

__device__ __forceinline__ float lerpf(float a, float b, float w) {
    // a*(1-w) + b*w, fused form
    return fmaf(w, b - a, a);
}

__global__ __launch_bounds__(256)
void hashgrid_embed_kernel(const float* __restrict__ xin,
                           const float* __restrict__ tables,
                           float* __restrict__ out,
                           int n)
{
    // res_l = floor(16 * (2^(1/3))^l), l = 0..7  (b = exp(ln(512/16)/15) = 2^(1/3))
    constexpr float RES[NLV_OUT] = {16.f, 20.f, 25.f, 32.f, 40.f, 50.f, 64.f, 80.f};

    const int i = blockIdx.x * blockDim.x + threadIdx.x;
    if (i >= n) return;

    // Coordinate load: three consecutive floats -> merges to global_load_b96.
    const float px = xin[3 * i + 0];
    const float py = xin[3 * i + 1];
    const float pz = xin[3 * i + 2];
    const float cx = fminf(fmaxf(px, 0.f), 1.f);
    const float cy = fminf(fmaxf(py, 0.f), 1.f);
    const float cz = fminf(fmaxf(pz, 0.f), 1.f);

    float o[16];

    #pragma unroll
    for (int l = 0; l < NLV_OUT; ++l) {
        const float res  = RES[l];
        const float grid = 1.0f / RES[l];          // compile-time constant

        // Base cell + interpolation weights (weights use the unclipped coord,
        // matching the reference's w = (x - vmin)/(vmax - vmin)).
        const float fbx = floorf(cx * res);
        const float fby = floorf(cy * res);
        const float fbz = floorf(cz * res);
        const float w0 = (px - fbx * grid) * res;
        const float w1 = (py - fby * grid) * res;
        const float w2 = (pz - fbz * grid) * res;

        const unsigned bx = (unsigned)(int)fbx;
        const unsigned by = (unsigned)(int)fby;
        const unsigned bz = (unsigned)(int)fbz;

        // Spatial hash, PRIMES[0] == 1.
        const unsigned hx0 = bx;
        const unsigned hx1 = bx + 1u;
        const unsigned hy0 = by * P1;
        const unsigned hy1 = hy0 + P1;
        const unsigned hz0 = bz * P2;
        const unsigned hz1 = hz0 + P2;

        const float2* __restrict__ tl =
            reinterpret_cast<const float2*>(tables) + (size_t)l * TSIZE;

        // 8 independent gathers (global_load_b64, L2-resident); the scheduler
        // batches them in an s_clause and interleaves across levels as
        // register pressure allows.
        const float2 e000 = tl[(hx0 ^ hy0 ^ hz0) & TMASK];  // corner (i=0,j=0,k=0)
        const float2 e001 = tl[(hx0 ^ hy0 ^ hz1) & TMASK];  // (0,0,1)
        const float2 e010 = tl[(hx0 ^ hy1 ^ hz0) & TMASK];  // (0,1,0)
        const float2 e011 = tl[(hx0 ^ hy1 ^ hz1) & TMASK];  // (0,1,1)
        const float2 e100 = tl[(hx1 ^ hy0 ^ hz0) & TMASK];  // (1,0,0)
        const float2 e101 = tl[(hx1 ^ hy0 ^ hz1) & TMASK];  // (1,0,1)
        const float2 e110 = tl[(hx1 ^ hy1 ^ hz0) & TMASK];  // (1,1,0)
        const float2 e111 = tl[(hx1 ^ hy1 ^ hz1) & TMASK];  // (1,1,1)

        // Trilinear interpolation, same reduction order as the reference:
        // x-axis (w0), then y-axis (w1), then z-axis (w2).
        const float c00x = lerpf(e000.x, e100.x, w0), c00y = lerpf(e000.y, e100.y, w0);
        const float c01x = lerpf(e001.x, e101.x, w0), c01y = lerpf(e001.y, e101.y, w0);
        const float c10x = lerpf(e010.x, e110.x, w0), c10y = lerpf(e010.y, e110.y, w0);
        const float c11x = lerpf(e011.x, e111.x, w0), c11y = lerpf(e011.y, e111.y, w0);
        const float c0x = lerpf(c00x, c10x, w1), c0y = lerpf(c00y, c10y, w1);
        const float c1x = lerpf(c01x, c11x, w1), c1y = lerpf(c01y, c11y, w1);
        o[2 * l + 0] = lerpf(c0x, c1x, w2);
        o[2 * l + 1] = lerpf(c0y, c1y, w2);
    }

    // 16 contiguous floats per point (64 B, 64 B-aligned): 4 x non-temporal
    // global_store_b128 so the streamed output never evicts tables from L2.
    v4f* op = reinterpret_cast<v4f*>(out + (size_t)i * 16);
    #pragma unroll
    for (int k = 0; k < 4; ++k) {
        v4f v = { o[4 * k + 0], o[4 * k + 1], o[4 * k + 2], o[4 * k + 3] };
        __builtin_nontemporal_store(v, op + k);
    }
}

extern "C" void kernel_launch(void* const* d_in, const int* in_sizes, int n_in,
                              void* d_out, int out_size, void* d_ws, size_t ws_size,
                              hipStream_t stream) {
    const float* x      = (const float*)d_in[0];   // (N_POINTS, 3) float32
    const float* tables = (const float*)d_in[1];   // (16, 2^19, 2) float32
    float* out          = (float*)d_out;           // (N_POINTS, 16) float32

    const int n = in_sizes[0] / 3;
    const int block = 256;
    const int grid = (n + block - 1) / block;
    hashgrid_embed_kernel<<<grid, block, 0, stream>>>(x, tables, out, n);
}